// ImpalaVoxelConvBlock_47132971107236
// MI455X (gfx1250) — compile-verified
//
#include <hip/hip_runtime.h>
#include <hip/hip_bf16.h>

// ---------------------------------------------------------------------------
// Sparse voxel conv (27-offset gather-GEMM) + batch norm, fp32 WMMA path.
//   feats: [N,32] f32   W: [27,32,32] f32   gamma/beta: [32]   neigh: [27,N] i32
// One wave32 computes TWO 16x32 output tiles with V_WMMA_F32_16X16X4_F32:
// B fragments are shared across the tile pair, and 4 independent accumulator
// chains let WMMAs overlap the gather loads. Sentinel indices (== N) are
// redirected branchlessly to a zeroed pad row in d_ws (no EXEC masking).
// ---------------------------------------------------------------------------

#define NPTS    1000000
#define NTILES  62500          // NPTS / 16 (exact)
#define NPAIRS  31250          // NTILES / 2 (exact)
#define K3      27
#define WAVES_PER_BLOCK 8

typedef __attribute__((ext_vector_type(2))) float v2f;
typedef __attribute__((ext_vector_type(8))) float v8f;

__device__ __forceinline__ float swap16(float x) {
    // ds_swizzle SWAPX16: new_lane = lane ^ 16  (offset = xor 0x10 << 10 | and 0x1f)
    return __int_as_float(__builtin_amdgcn_ds_swizzle(__float_as_int(x), 0x401F));
}

// ---------------------------------------------------------------------------
// Pack W[k][cin][cout] into per-lane B-operand fragments:
//   WP2[((k*2 + t)*8 + c)*32 + lane] = (W[k][4c+2h][16t+n], W[k][4c+2h+1][16t+n])
// with h = lane>>4, n = lane&15.  (B fragment for K-chunk c, COUT-tile t.)
// ---------------------------------------------------------------------------
__global__ void pack_w_kernel(const float* __restrict__ W, float2* __restrict__ WP2) {
    int tid = blockIdx.x * blockDim.x + threadIdx.x;   // 13824 fragments total
    if (tid >= K3 * 2 * 8 * 32) return;
    int lane = tid & 31;
    int c    = (tid >> 5) & 7;
    int t    = (tid >> 8) & 1;
    int k    = tid >> 9;
    int h    = lane >> 4;
    int n    = lane & 15;
    int cin0 = 4 * c + 2 * h;
    int cout = 16 * t + n;
    float2 v;
    v.x = W[(k * 32 + cin0) * 32 + cout];
    v.y = W[(k * 32 + cin0 + 1) * 32 + cout];
    WP2[tid] = v;
}

// ---------------------------------------------------------------------------
// Main conv: one wave per PAIR of 16-point tiles; four 16x16 f32 accumulators.
// Also accumulates per-channel sum / sum-of-squares for the batch norm.
// gsum layout: [0..31] = sum(out[:,c]),  [32..63] = sum(out[:,c]^2)
// ---------------------------------------------------------------------------
__global__ void __launch_bounds__(256)
conv_kernel(const float* __restrict__ F, const int* __restrict__ NI,
            const float2* __restrict__ WP2, const float2* __restrict__ ZROW,
            float* __restrict__ out, float* __restrict__ gsum) {
    __shared__ float ssum[32];
    __shared__ float ssq[32];
    const int tid = threadIdx.x;
    if (tid < 32) { ssum[tid] = 0.0f; ssq[tid] = 0.0f; }
    __syncthreads();

    const int wave = tid >> 5;
    const int lane = tid & 31;
    const int pair = blockIdx.x * WAVES_PER_BLOCK + wave;

    if (pair < NPAIRS) {            // wave-uniform guard: EXEC all-1s inside
        const int m  = lane & 15;   // A row (M = m) / C-D column (N = m)
        const int h  = lane >> 4;   // lane half
        const int t0 = pair * 2;    // first tile of the pair
        const int p0 = t0 * 16 + m;
        const int p1 = p0 + 16;

        v8f accA0 = {}, accA1 = {};   // tile0: COUT 0..15 / 16..31
        v8f accB0 = {}, accB1 = {};   // tile1: COUT 0..15 / 16..31
        const float2* F2 = (const float2*)F;
        const float2* Z2 = ZROW + h;  // zeroed pad row (sentinel target)

        for (int k = 0; k < K3; ++k) {
            const int idxA = NI[k * NPTS + p0];
            const int idxB = NI[k * NPTS + p1];
            // Branchless feats_pad: select the pointer, not the load.
            const float2* rowA = (idxA < NPTS) ? (F2 + (size_t)idxA * 16 + h) : Z2;
            const float2* rowB = (idxB < NPTS) ? (F2 + (size_t)idxB * 16 + h) : Z2;
            const float2* wb   = WP2 + (size_t)k * 512 + lane;
#pragma unroll
            for (int c = 0; c < 8; ++c) {               // K chunks of 4
                float2 avA = rowA[2 * c];
                float2 avB = rowB[2 * c];
                float2 b0v = wb[c * 32];                // t = 0 fragment
                float2 b1v = wb[256 + c * 32];          // t = 1 fragment
                v2f aA; aA.x = avA.x; aA.y = avA.y;
                v2f aB; aB.x = avB.x; aB.y = avB.y;
                v2f b0; b0.x = b0v.x; b0.y = b0v.y;
                v2f b1; b1.x = b1v.x; b1.y = b1v.y;
                accA0 = __builtin_amdgcn_wmma_f32_16x16x4_f32(
                        false, aA, false, b0, (short)0, accA0, false, false);
                accA1 = __builtin_amdgcn_wmma_f32_16x16x4_f32(
                        false, aA, false, b1, (short)0, accA1, false, false);
                accB0 = __builtin_amdgcn_wmma_f32_16x16x4_f32(
                        false, aB, false, b0, (short)0, accB0, false, false);
                accB1 = __builtin_amdgcn_wmma_f32_16x16x4_f32(
                        false, aB, false, b1, (short)0, accB1, false, false);
            }
        }

        // Store pre-BN tiles + fold per-lane column sums (both tiles).
        float s0 = 0.0f, q0 = 0.0f, s1 = 0.0f, q1 = 0.0f;
#pragma unroll
        for (int r = 0; r < 8; ++r) {
            const int rowT0 = t0 * 16 + r + 8 * h;       // C/D layout: M = r + 8h
            const int rowT1 = rowT0 + 16;
            out[(size_t)rowT0 * 32 + m]      = accA0[r];
            out[(size_t)rowT0 * 32 + 16 + m] = accA1[r];
            out[(size_t)rowT1 * 32 + m]      = accB0[r];
            out[(size_t)rowT1 * 32 + 16 + m] = accB1[r];
            s0 += accA0[r] + accB0[r];
            q0 += accA0[r] * accA0[r] + accB0[r] * accB0[r];
            s1 += accA1[r] + accB1[r];
            q1 += accA1[r] * accA1[r] + accB1[r] * accB1[r];
        }
        // Combine the two lane halves (rows 0-7 vs 8-15 of each column).
        s0 += swap16(s0); q0 += swap16(q0);
        s1 += swap16(s1); q1 += swap16(q1);
        if (h == 0) {
            atomicAdd(&ssum[m], s0);      atomicAdd(&ssq[m], q0);
            atomicAdd(&ssum[16 + m], s1); atomicAdd(&ssq[16 + m], q1);
        }
    }

    __syncthreads();
    if (tid < 64) {
        const float v = (tid < 32) ? ssum[tid] : ssq[tid - 32];
        atomicAdd(&gsum[tid], v);
    }
}

// ---------------------------------------------------------------------------
// Batch-norm finalize (in place on d_out): biased stats, eps = 1e-5.
// ---------------------------------------------------------------------------
__global__ void bn_kernel(float* __restrict__ out, const float* __restrict__ gsum,
                          const float* __restrict__ gamma, const float* __restrict__ beta) {
    const size_t i = (size_t)blockIdx.x * blockDim.x + threadIdx.x;
    if (i >= (size_t)NPTS * 32) return;
    const int c = (int)(i & 31);
    const float invN = 1.0f / (float)NPTS;
    const float mean = gsum[c] * invN;
    const float var  = gsum[32 + c] * invN - mean * mean;
    const float inv  = rsqrtf(var + 1e-5f);
    out[i] = (out[i] - mean) * inv * gamma[c] + beta[c];
}

extern "C" void kernel_launch(void* const* d_in, const int* in_sizes, int n_in,
                              void* d_out, int out_size, void* d_ws, size_t ws_size,
                              hipStream_t stream) {
    const float* feats = (const float*)d_in[0];
    const float* W     = (const float*)d_in[1];
    const float* gamma = (const float*)d_in[2];
    const float* beta  = (const float*)d_in[3];
    const int*   neigh = (const int*)d_in[4];
    float* out = (float*)d_out;

    // d_ws layout: [0,512)   BN reduction slots (64 f32 + pad)
    //              [512,640) zeroed pad row (sentinel gather target, 32 f32)
    //              [1024,..) packed weights (108 KB, stays L2-hot)
    float*        gsum = (float*)d_ws;
    const float2* zrow = (const float2*)((char*)d_ws + 512);
    float2*       WP2  = (float2*)((char*)d_ws + 1024);

    (void)in_sizes; (void)n_in; (void)out_size; (void)ws_size;

    hipMemsetAsync(d_ws, 0, 1024, stream);              // zero sums + pad row

    pack_w_kernel<<<54, 256, 0, stream>>>(W, WP2);      // 54*256 = 13824 frags

    const int blocks = (NPAIRS + WAVES_PER_BLOCK - 1) / WAVES_PER_BLOCK;  // 3907
    conv_kernel<<<blocks, 256, 0, stream>>>(feats, neigh, WP2, zrow, out, gsum);

    const size_t tot = (size_t)NPTS * 32;
    bn_kernel<<<(unsigned)((tot + 255) / 256), 256, 0, stream>>>(out, gsum, gamma, beta);
}